// MILLoss_45681272160447
// MI455X (gfx1250) — compile-verified
//
#include <hip/hip_runtime.h>

#define MIL_EPS 1e-7f
#define NUM_BAGS 65536
#define MASK_WORDS (NUM_BAGS / 32)   // 2048

typedef __attribute__((ext_vector_type(2))) float v2f;
typedef __attribute__((ext_vector_type(8))) float v8f;

// ---------------------------------------------------------------------------
// Workspace layout (d_ws):
//   float    pos_seg [NUM_BAGS]     -- segment sum of log(1-p+eps) over positives
//   unsigned negMask [MASK_WORDS]   -- bit b set => bag b has >=1 negative instance
//   unsigned posMask [MASK_WORDS]   -- bit b set => bag b has >=1 positive instance
//   float    scalars [4]            -- [0]=neg_sum [1]=num_neg [2]=num_pos [3]=per_bag_sum
// ---------------------------------------------------------------------------

__global__ void mil_zero_ws(unsigned* __restrict__ ws, int nwords) {
    int i = blockIdx.x * blockDim.x + threadIdx.x;
    if (i < nwords) ws[i] = 0u;
}

// Pass 1: stream all N instances (memory-bound: 96 MB @ 23.3 TB/s ~ 4 us floor).
__global__ void mil_main(const float* __restrict__ probas,
                         const int*   __restrict__ labels,
                         const int*   __restrict__ bag_ids,
                         float*       __restrict__ pos_seg,
                         unsigned*    __restrict__ negMask,
                         unsigned*    __restrict__ posMask,
                         float*       __restrict__ scalars,
                         int n) {
    int idx    = blockIdx.x * blockDim.x + threadIdx.x;
    int stride = gridDim.x * blockDim.x;
    float local_neg = 0.0f;
    for (int i = idx; i < n; i += stride) {
        float p   = probas[i];
        int   lab = labels[i];
        int   bag = bag_ids[i];
        float ln  = logf(1.0f - p + MIL_EPS);
        unsigned w = ((unsigned)bag) >> 5;
        unsigned b = 1u << (bag & 31);
        if (lab == 0) {
            local_neg += ln;
            if ((negMask[w] & b) == 0u) atomicOr(&negMask[w], b);   // mostly skipped once bit set
        } else {
            atomicAdd(&pos_seg[bag], ln);                           // L2-resident f32 atomic
            if ((posMask[w] & b) == 0u) atomicOr(&posMask[w], b);
        }
    }
    // wave32 shuffle reduction, one global f32 atomic per wave
    #pragma unroll
    for (int o = 16; o > 0; o >>= 1) local_neg += __shfl_xor(local_neg, o, 32);
    if ((threadIdx.x & 31) == 0) atomicAdd(&scalars[0], local_neg);
}

// Pass 2: 65536 bags -> 3 scalars, reduced on the matrix pipe.
// Each wave owns 256 bags; per step it packs 64 f32 values into a 16x4 A tile
// and accumulates C += A x ones(4x16) via V_WMMA_F32_16X16X4_F32 (exact f32).
// Launch MUST be 32 blocks x 256 threads (256 waves, fully converged: EXEC all-1s).
__global__ void mil_bag_reduce_wmma(const float*    __restrict__ pos_seg,
                                    const unsigned* __restrict__ negMask,
                                    const unsigned* __restrict__ posMask,
                                    float*          __restrict__ scalars) {
    const int lane = threadIdx.x & 31;
    const int wave = (blockIdx.x * blockDim.x + threadIdx.x) >> 5;   // 0..255
    const int base = wave * 256;

    v8f cV = {};  // sum of per-bag positive loss terms
    v8f cN = {};  // count of bags with a negative
    v8f cP = {};  // count of bags with a positive
    v2f ones; ones.x = 1.0f; ones.y = 1.0f;   // B = all-ones 4x16

    #pragma unroll
    for (int it = 0; it < 4; ++it) {
        float vv[2], nn[2], pp[2];
        #pragma unroll
        for (int t = 0; t < 2; ++t) {
            int i = base + it * 64 + lane + t * 32;
            unsigned nb = (negMask[i >> 5] >> (i & 31)) & 1u;
            unsigned pb = (posMask[i >> 5] >> (i & 31)) & 1u;
            float s   = pos_seg[i];
            float anp = fminf(expf(s), 1.0f);            // clamp(exp(seg), max=1)
            float lv  = logf(1.0f - anp + MIL_EPS);
            vv[t] = pb ? lv : 0.0f;                      // branchless: keeps EXEC uniform
            nn[t] = (float)nb;
            pp[t] = (float)pb;
        }
        v2f aV; aV.x = vv[0]; aV.y = vv[1];
        v2f aN; aN.x = nn[0]; aN.y = nn[1];
        v2f aP; aP.x = pp[0]; aP.y = pp[1];
        cV = __builtin_amdgcn_wmma_f32_16x16x4_f32(false, aV, false, ones, (short)0, cV, false, false);
        cN = __builtin_amdgcn_wmma_f32_16x16x4_f32(false, aN, false, ones, (short)0, cN, false, false);
        cP = __builtin_amdgcn_wmma_f32_16x16x4_f32(false, aP, false, ones, (short)0, cP, false, false);
    }

    // D[m][n] identical across columns; fold 8 VGPRs (rows M=0..7 / 8..15 per half)
    // then add the opposite 16-lane half: every lane ends with the full wave total.
    float tV = 0.0f, tN = 0.0f, tP = 0.0f;
    #pragma unroll
    for (int k = 0; k < 8; ++k) { tV += cV[k]; tN += cN[k]; tP += cP[k]; }
    tV += __shfl_xor(tV, 16, 32);
    tN += __shfl_xor(tN, 16, 32);
    tP += __shfl_xor(tP, 16, 32);

    if (lane == 0) {
        atomicAdd(&scalars[3], tV);
        atomicAdd(&scalars[1], tN);
        atomicAdd(&scalars[2], tP);
    }
}

__global__ void mil_finalize(const float* __restrict__ scalars,
                             float* __restrict__ out) {
    float neg_sum = scalars[0];
    float num_neg = scalars[1];
    float num_pos = scalars[2];
    float pb_sum  = scalars[3];
    float neg_loss = (num_neg > 0.0f) ? (-neg_sum / fmaxf(num_neg, 1.0f)) : 0.0f;
    float pos_loss = (num_pos > 0.0f) ? (-pb_sum  / fmaxf(num_pos, 1.0f)) : 0.0f;
    out[0] = neg_loss + pos_loss;
}

extern "C" void kernel_launch(void* const* d_in, const int* in_sizes, int n_in,
                              void* d_out, int out_size, void* d_ws, size_t ws_size,
                              hipStream_t stream) {
    (void)n_in; (void)out_size; (void)ws_size;

    const float* probas  = (const float*)d_in[0];
    const int*   labels  = (const int*)d_in[1];
    const int*   bag_ids = (const int*)d_in[2];
    const int    n       = in_sizes[0];

    float*    pos_seg = (float*)d_ws;
    unsigned* negMask = (unsigned*)(pos_seg + NUM_BAGS);
    unsigned* posMask = negMask + MASK_WORDS;
    float*    scalars = (float*)(posMask + MASK_WORDS);

    const int nwords = NUM_BAGS + 2 * MASK_WORDS + 4;
    mil_zero_ws<<<(nwords + 255) / 256, 256, 0, stream>>>((unsigned*)d_ws, nwords);

    mil_main<<<4096, 256, 0, stream>>>(probas, labels, bag_ids,
                                       pos_seg, negMask, posMask, scalars, n);

    // exactly 256 waves: 256 bags each -> covers NUM_BAGS with no divergence
    mil_bag_reduce_wmma<<<32, 256, 0, stream>>>(pos_seg, negMask, posMask, scalars);

    mil_finalize<<<1, 1, 0, stream>>>(scalars, (float*)d_out);
}